// AdaFastFoodMergedModel_50568944943730
// MI455X (gfx1250) — compile-verified
//
#include <hip/hip_runtime.h>
#include <cmath>
#include <cstdint>

typedef float v2f __attribute__((ext_vector_type(2)));
typedef float v8f __attribute__((ext_vector_type(8)));

#define NROW 2048          // FWHT length handled per pass (2048 = 2^11; 2^22 total)
#define ROWS_PER_BLK 8
#define COLS_PER_BLK 16
#define CPAD 17            // LDS row pad (17 coprime with 64 banks -> conflict-free)

// ---------- CDNA5 async global<->LDS helpers (ASYNCcnt path, ISA ch.10/15.18) ----------

__device__ __forceinline__ void async_load_b128(const void* gaddr, void* lds) {
    unsigned l = (unsigned)(uintptr_t)lds;                 // LDS byte offset (addr[31:0])
    unsigned long long g = (unsigned long long)(uintptr_t)gaddr;
    asm volatile("global_load_async_to_lds_b128 %0, %1, off" :: "v"(l), "v"(g) : "memory");
}
__device__ __forceinline__ void async_load_b32(const void* gaddr, void* lds) {
    unsigned l = (unsigned)(uintptr_t)lds;
    unsigned long long g = (unsigned long long)(uintptr_t)gaddr;
    asm volatile("global_load_async_to_lds_b32 %0, %1, off" :: "v"(l), "v"(g) : "memory");
}
__device__ __forceinline__ void async_store_b32(void* gaddr, const void* lds) {
    unsigned l = (unsigned)(uintptr_t)lds;
    unsigned long long g = (unsigned long long)(uintptr_t)gaddr;
    asm volatile("global_store_async_from_lds_b32 %0, %1, off" :: "v"(g), "v"(l) : "memory");
}
__device__ __forceinline__ void wait_async0() {
    asm volatile("s_wait_asynccnt 0" ::: "memory");
}

// Sylvester Hadamard sign H16[m][k] = (-1)^popcount(m&k)
__device__ __forceinline__ float hsign16(int m, int k) {
    return (__popc(m & k) & 1) ? -1.0f : 1.0f;
}

// Y(16x16) = H16 * X(16x16), X/Y live in LDS.
// Element (i, n) of X is at sm[jbase_n + i*strideL] where jbase_n is this lane's
// column base (lane's n == mrow per the CDNA5 B/D layout). Four chained
// V_WMMA_F32_16X16X4_F32 ops cover K=16. A operand (H16 columns) is computed
// in-register from popcount; exact in f32 since entries are +-1.
__device__ __forceinline__ void hadamard16_wmma(float* sm, int jbase, int strideL,
                                                int half, int mrow) {
    v8f acc = {0.f, 0.f, 0.f, 0.f, 0.f, 0.f, 0.f, 0.f};
#pragma unroll
    for (int ch = 0; ch < 4; ++ch) {
        const int k0 = ch * 4 + half * 2;      // A/B K-pair for this lane half
        v2f a, b;
        a.x = hsign16(mrow, k0);
        a.y = hsign16(mrow, k0 + 1);
        b.x = sm[jbase + k0 * strideL];
        b.y = sm[jbase + (k0 + 1) * strideL];
        acc = __builtin_amdgcn_wmma_f32_16x16x4_f32(
            /*neg_a=*/false, a, /*neg_b=*/false, b,
            /*c_mod=*/(short)0, acc, /*reuse_a=*/false, /*reuse_b=*/false);
    }
#pragma unroll
    for (int v = 0; v < 8; ++v)
        sm[jbase + (v + half * 8) * strideL] = acc[v];   // D row m = v + half*8, col n = mrow
}

// In-LDS orthogonal (unscaled) FWHT of 2048 elements, one wave.
// Element e of the vector lives at sm[CB + e*ES].
// Decomposition: H2048 = H16(bits 10:7) x H16(bits 6:3) x H8(bits 2:0).
__device__ void fwht2048_wave(float* sm, int ES, int CB) {
    const int lane = threadIdx.x & 31;
    const int half = lane >> 4;
    const int mrow = lane & 15;

    // --- radix-8 stage on bits [2:0], in registers (256 groups, 8/lane) ---
#pragma unroll 1
    for (int it = 0; it < 8; ++it) {
        const int g = lane + it * 32;
        const int base = CB + g * 8 * ES;
        float r[8];
#pragma unroll
        for (int j = 0; j < 8; ++j) r[j] = sm[base + j * ES];
#pragma unroll
        for (int h = 1; h < 8; h <<= 1)
#pragma unroll
            for (int i0 = 0; i0 < 8; i0 += 2 * h)
#pragma unroll
                for (int j = 0; j < h; ++j) {
                    const float u = r[i0 + j], v = r[i0 + j + h];
                    r[i0 + j] = u + v;
                    r[i0 + j + h] = u - v;
                }
#pragma unroll
        for (int j = 0; j < 8; ++j) sm[base + j * ES] = r[j];
    }
    __builtin_amdgcn_wave_barrier();   // cross-lane LDS results consumed below

    // --- radix-16 on bits [6:3] (stride 8): 8 tiles of 16 vectors ---
#pragma unroll 1
    for (int t = 0; t < 8; ++t) {
        // tile columns n -> (a = 2t + n>>3, q = n&7); element e = a*128 + p*8 + q
        const int jo = (2 * t + (mrow >> 3)) * 128 + (mrow & 7);
        hadamard16_wmma(sm, CB + jo * ES, 8 * ES, half, mrow);
    }
    __builtin_amdgcn_wave_barrier();

    // --- radix-16 on bits [10:7] (stride 128): 8 tiles of 16 vectors ---
#pragma unroll 1
    for (int u = 0; u < 8; ++u) {
        const int jo = u * 16 + mrow;          // pq = u*16 + n
        hadamard16_wmma(sm, CB + jo * ES, 128 * ES, half, mrow);
    }
}

// ---------- FWHT pass kernels ----------

// Row pass: FWHT-2048 over each contiguous 2048-block. 8 rows per block, wave w owns row w.
// Loads arrive via CDNA5 async DMA (global -> LDS, ASYNCcnt); scaled stores via float4.
__global__ void __launch_bounds__(256) k_fwht_rows(const float* __restrict__ in,
                                                   float* __restrict__ out, float scale) {
    __shared__ float sm[ROWS_PER_BLK * NROW];          // 64 KB
    const size_t base = (size_t)blockIdx.x * (ROWS_PER_BLK * NROW);
    // 16B/lane async copies; both global and LDS addresses are 16B aligned.
#pragma unroll 1
    for (int i = threadIdx.x * 4; i < ROWS_PER_BLK * NROW; i += 1024)
        async_load_b128(in + base + i, &sm[i]);
    wait_async0();
    __syncthreads();
    fwht2048_wave(sm, 1, (threadIdx.x >> 5) * NROW);
    __syncthreads();
#pragma unroll 1
    for (int i = threadIdx.x * 4; i < ROWS_PER_BLK * NROW; i += 1024) {
        float4 v;
        v.x = sm[i + 0] * scale;
        v.y = sm[i + 1] * scale;
        v.z = sm[i + 2] * scale;
        v.w = sm[i + 3] * scale;
        *(float4*)(out + base + i) = v;
    }
}

// Col pass: FWHT-2048 at stride `ncols` (matrix columns). 16 columns per block,
// padded LDS tile; wave w owns columns 2w, 2w+1. Pure copy in/out -> both
// directions use the async LDS DMA path (b32 since CPAD breaks 16B alignment).
__global__ void __launch_bounds__(256) k_fwht_cols(const float* __restrict__ in,
                                                   float* __restrict__ out, int ncols) {
    __shared__ float sm[NROW * CPAD];                  // 139264 B (< 320 KB WGP LDS)
    const int col0 = blockIdx.x * COLS_PER_BLK;
    const int c = threadIdx.x & 15;
    const int r0 = threadIdx.x >> 4;
#pragma unroll 1
    for (int r = r0; r < NROW; r += 16)
        async_load_b32(in + (size_t)r * ncols + col0 + c, &sm[r * CPAD + c]);
    wait_async0();
    __syncthreads();
    const int w = threadIdx.x >> 5;
    fwht2048_wave(sm, CPAD, 2 * w);
    fwht2048_wave(sm, CPAD, 2 * w + 1);
    __syncthreads();
#pragma unroll 1
    for (int r = r0; r < NROW; r += 16)
        async_store_b32(out + (size_t)r * ncols + col0 + c, &sm[r * CPAD + c]);
    wait_async0();   // (S_ENDPGM also waits-idle; explicit for clarity)
}

// ---------- Elementwise / permutation kernels ----------

__global__ void k_combine(const float* __restrict__ tv, const float* __restrict__ w,
                          const float* __restrict__ bv, float* __restrict__ out,
                          size_t L, int T) {
    const size_t i = (size_t)blockIdx.x * blockDim.x + threadIdx.x;
    if (i >= L) return;
    float s = 0.f;
    for (int t = 0; t < T; ++t) s += w[t] * tv[(size_t)t * L + i];
    out[i] = s * bv[i];
}

__global__ void k_gather_dstmul(const float* __restrict__ in, const int* __restrict__ idx,
                                const float* __restrict__ g, float* __restrict__ out, size_t L) {
    const size_t i = (size_t)blockIdx.x * blockDim.x + threadIdx.x;
    if (i >= L) return;
    out[i] = in[idx[i]] * g[i];                         // x[Pi]*G
}

__global__ void k_gather_srcmul(const float* __restrict__ in, const int* __restrict__ idx,
                                const float* __restrict__ g, float* __restrict__ out, size_t L) {
    const size_t i = (size_t)blockIdx.x * blockDim.x + threadIdx.x;
    if (i >= L) return;
    const int q = idx[i];
    out[i] = in[q] * g[q];                              // (fwht(y)*G)[inv_Pi]
}

__global__ void k_zero(float* __restrict__ out, size_t L) {
    const size_t i = (size_t)blockIdx.x * blockDim.x + threadIdx.x;
    if (i < L) out[i] = 0.f;
}

__global__ void k_scatter_sel(const float* __restrict__ src, const int* __restrict__ ridx,
                              float* __restrict__ dst, int m) {
    const int j = blockIdx.x * blockDim.x + threadIdx.x;
    if (j >= m) return;
    const int p = ridx[j];
    dst[p] = src[p];                                    // y_full[row_idx] = x2[row_idx]
}

__global__ void k_final(const float* __restrict__ pre, const float* __restrict__ bv,
                        float* __restrict__ out, size_t L) {
    const size_t i = (size_t)blockIdx.x * blockDim.x + threadIdx.x;
    if (i >= L) return;
    out[i] = pre[i] + out[i] * bv[i];
}

extern "C" void kernel_launch(void* const* d_in, const int* in_sizes, int n_in,
                              void* d_out, int out_size, void* d_ws, size_t ws_size,
                              hipStream_t stream) {
    const float* tv  = (const float*)d_in[0];   // task_vectors (T,L)
    const float* aw  = (const float*)d_in[1];   // ada_weights  (T)
    const float* pre = (const float*)d_in[2];   // pretrained   (L)
    const float* Bv  = (const float*)d_in[3];   // B            (L)
    const float* G   = (const float*)d_in[4];   // G            (L)
    const int*   Pi  = (const int*)d_in[5];     // Pi           (L)
    const int*   iPi = (const int*)d_in[6];     // inv_Pi       (L)
    const int*   rix = (const int*)d_in[7];     // row_idx      (m)

    const int    T = in_sizes[1];
    const size_t L = (size_t)in_sizes[3];       // 2^22
    const int    m = in_sizes[7];
    const int    ncols = (int)(L / NROW);       // 2048
    const float  invs  = 1.0f / sqrtf((float)L);  // orthonormal factor, applied in row pass

    float* A = (float*)d_ws;    // 16 MB scratch
    float* O = (float*)d_out;   // used as second scratch, final result in-place

    const int EB = (int)((L + 255) / 256);      // elementwise grid
    const int RG = (int)(L / (ROWS_PER_BLK * NROW));   // row-pass grid (256)
    const int CG = ncols / COLS_PER_BLK;               // col-pass grid (128)
    const int MB = (m + 255) / 256;

    // x = B .* (sum_t w_t * tv_t)          [linearity folds T=8 -> 1]
    k_combine<<<EB, 256, 0, stream>>>(tv, aw, Bv, A, L, T);
    // FWHT #1  (1/sqrt(L) folded into row pass)
    k_fwht_rows<<<RG, 256, 0, stream>>>(A, O, invs);
    k_fwht_cols<<<CG, 256, 0, stream>>>(O, A, ncols);
    // x2 = x[Pi] * G
    k_gather_dstmul<<<EB, 256, 0, stream>>>(A, Pi, G, O, L);
    // FWHT #2
    k_fwht_rows<<<RG, 256, 0, stream>>>(O, A, invs);
    k_fwht_cols<<<CG, 256, 0, stream>>>(A, O, ncols);
    // y_full = mask(row_idx) .* x2        [scale * 1/scale cancels exactly]
    k_zero<<<EB, 256, 0, stream>>>(A, L);
    k_scatter_sel<<<MB, 256, 0, stream>>>(O, rix, A, m);
    // FWHT #3
    k_fwht_rows<<<RG, 256, 0, stream>>>(A, O, invs);
    k_fwht_cols<<<CG, 256, 0, stream>>>(O, A, ncols);
    // (fwht(y_full)*G)[inv_Pi]
    k_gather_srcmul<<<EB, 256, 0, stream>>>(A, iPi, G, O, L);
    // FWHT #4
    k_fwht_rows<<<RG, 256, 0, stream>>>(O, A, invs);
    k_fwht_cols<<<CG, 256, 0, stream>>>(A, O, ncols);
    // out = pretrained + y_full .* B   (in place on d_out)
    k_final<<<EB, 256, 0, stream>>>(pre, Bv, O, L);
}